// My_RGCN_53455162966646
// MI455X (gfx1250) — compile-verified
//
#include <hip/hip_runtime.h>
#include <hip/hip_bf16.h>

typedef float v2f __attribute__((ext_vector_type(2)));
typedef float v8f __attribute__((ext_vector_type(8)));

#define NODES 50000
#define EDGES 800000
#define FIN   768
#define FHID  256

// ---------------------------------------------------------------------------
// small utility kernels
// ---------------------------------------------------------------------------
__global__ void fill_zero_kernel(float* __restrict__ p, int n) {
    int i = blockIdx.x * blockDim.x + threadIdx.x;
    if (i < n) p[i] = 0.0f;
}

__global__ void count_edges_kernel(const int* __restrict__ dst,
                                   const int* __restrict__ et,
                                   float* __restrict__ cnt, int E, int Nn) {
    int e = blockIdx.x * blockDim.x + threadIdx.x;
    if (e < E) atomicAdd(&cnt[et[e] * Nn + dst[e]], 1.0f);
}

__global__ void invert_cnt_kernel(float* __restrict__ cnt, int n) {
    int i = blockIdx.x * blockDim.x + threadIdx.x;
    if (i < n) cnt[i] = 1.0f / fmaxf(cnt[i], 1.0f);
}

// ---------------------------------------------------------------------------
// fp32 WMMA GEMM:  C[M,N] = act(A[M,K] @ W[K,N] + bias)
// One wave -> 16(M) x 64(N) tile, 4 accumulators, V_WMMA_F32_16X16X4_F32.
// A-fragment layout (ISA 7.12.2, 32-bit A 16x4): lanes 0-15 hold K={0,1},
// lanes 16-31 hold K={2,3}; each lane's 2 values are K-contiguous -> float2.
// B-fragment: row (K) striped across VGPRs, N striped across lanes.
// Requires M%16==0, N%64==0, K%4==0 (true for all layers here).
// ---------------------------------------------------------------------------
__global__ __launch_bounds__(256)
void wmma_gemm_f32(const float* __restrict__ A, const float* __restrict__ W,
                   const float* __restrict__ bias, float* __restrict__ C,
                   int M, int K, int N, int leaky) {
    const int wave = blockIdx.x * (blockDim.x >> 5) + (threadIdx.x >> 5);
    const int ntx  = N >> 6;       // 64-wide N tiles
    const int nty  = M >> 4;       // 16-tall M tiles
    if (wave >= ntx * nty) return;
    const int ty = wave / ntx, tx = wave - ty * ntx;
    const int m0 = ty << 4, n0 = tx << 6;
    const int lane = threadIdx.x & 31;
    const int l15  = lane & 15;
    const int hi   = lane >> 4;    // 0: K={0,1}, 1: K={2,3}

    v8f acc0 = {}, acc1 = {}, acc2 = {}, acc3 = {};
    const float* arow = A + (size_t)(m0 + l15) * K + 2 * hi;

    for (int k0 = 0; k0 < K; k0 += 4) {
        v2f a = *(const v2f*)(arow + k0);
        const float* bp = W + (size_t)(k0 + 2 * hi) * N + n0 + l15;
        v2f b0, b1, b2, b3;
        b0.x = bp[0];   b0.y = bp[N];
        b1.x = bp[16];  b1.y = bp[N + 16];
        b2.x = bp[32];  b2.y = bp[N + 32];
        b3.x = bp[48];  b3.y = bp[N + 48];
        acc0 = __builtin_amdgcn_wmma_f32_16x16x4_f32(false, a, false, b0, (short)0, acc0, false, false);
        acc1 = __builtin_amdgcn_wmma_f32_16x16x4_f32(false, a, false, b1, (short)0, acc1, false, false);
        acc2 = __builtin_amdgcn_wmma_f32_16x16x4_f32(false, a, false, b2, (short)0, acc2, false, false);
        acc3 = __builtin_amdgcn_wmma_f32_16x16x4_f32(false, a, false, b3, (short)0, acc3, false, false);
    }

    // Epilogue. C/D layout: VGPR r = row (m0 + r + 8*hi), col = n0 + (lane&15).
    const int colbase = n0 + l15;
#define RGCN_EPILOGUE(ACC, TOFF)                                          \
    do {                                                                  \
        float b_ = bias ? bias[colbase + (TOFF)] : 0.0f;                  \
        _Pragma("unroll")                                                 \
        for (int r = 0; r < 8; ++r) {                                     \
            float v = (ACC)[r] + b_;                                      \
            if (leaky) v = (v >= 0.0f) ? v : 0.01f * v;                   \
            C[(size_t)(m0 + r + 8 * hi) * N + colbase + (TOFF)] = v;      \
        }                                                                 \
    } while (0)
    RGCN_EPILOGUE(acc0, 0);
    RGCN_EPILOGUE(acc1, 16);
    RGCN_EPILOGUE(acc2, 32);
    RGCN_EPILOGUE(acc3, 48);
#undef RGCN_EPILOGUE
}

// ---------------------------------------------------------------------------
// Fused mean scatter: out[dst] += (1/max(cnt,1)) * h_{type}[src]
// One wave per edge, 8 fp32 atomics per lane (FHID=256).
// ---------------------------------------------------------------------------
__global__ __launch_bounds__(256)
void scatter_mean_kernel(const float* __restrict__ h0, const float* __restrict__ h1,
                         const int* __restrict__ src, const int* __restrict__ dstv,
                         const int* __restrict__ et, const float* __restrict__ cinv,
                         float* __restrict__ out, int E, int Nn) {
    long gid = (long)blockIdx.x * blockDim.x + threadIdx.x;
    int e = (int)(gid >> 5);
    if (e >= E) return;
    int lane = (int)(gid & 31);
    int s = src[e], d = dstv[e], t = et[e];
    float w = cinv[t * Nn + d];
    const float* hp = (t ? h1 : h0) + (size_t)s * FHID;
    float* op = out + (size_t)d * FHID;
#pragma unroll
    for (int j = 0; j < FHID / 32; ++j)
        atomicAdd(&op[lane + 32 * j], w * hp[lane + 32 * j]);
}

// ---------------------------------------------------------------------------
// Final [256 -> 2] projection + softmax. One wave per node, shuffle reduce.
// ---------------------------------------------------------------------------
__global__ __launch_bounds__(256)
void final_softmax_kernel(const float* __restrict__ h, const float* __restrict__ Wo,
                          const float* __restrict__ bo, float* __restrict__ out, int Nn) {
    int wave = blockIdx.x * (blockDim.x >> 5) + (threadIdx.x >> 5);
    if (wave >= Nn) return;
    int lane = threadIdx.x & 31;
    const float* hp = h + (size_t)wave * FHID;
    float a0 = 0.0f, a1 = 0.0f;
#pragma unroll
    for (int j = lane; j < FHID; j += 32) {
        float v = hp[j];
        a0 += v * Wo[2 * j];
        a1 += v * Wo[2 * j + 1];
    }
#pragma unroll
    for (int off = 16; off > 0; off >>= 1) {
        a0 += __shfl_xor(a0, off, 32);
        a1 += __shfl_xor(a1, off, 32);
    }
    if (lane == 0) {
        a0 += bo[0]; a1 += bo[1];
        float m = fmaxf(a0, a1);
        float e0 = __expf(a0 - m), e1 = __expf(a1 - m);
        float s = e0 + e1;
        out[2 * wave]     = e0 / s;
        out[2 * wave + 1] = e1 / s;
    }
}

// ---------------------------------------------------------------------------
extern "C" void kernel_launch(void* const* d_in, const int* in_sizes, int n_in,
                              void* d_out, int out_size, void* d_ws, size_t ws_size,
                              hipStream_t stream) {
    (void)in_sizes; (void)n_in; (void)out_size; (void)ws_size;

    const float* x     = (const float*)d_in[0];
    const int*   eidx  = (const int*)d_in[1];
    const int*   esrc  = eidx;
    const int*   edst  = eidx + EDGES;
    // d_in[2] = edge_weight: unused by the reference computation
    const int*   etype = (const int*)d_in[3];
    const float* fc1W  = (const float*)d_in[4];
    const float* fc1b  = (const float*)d_in[5];
    const float* r1w   = (const float*)d_in[6];
    const float* r1root= (const float*)d_in[7];
    const float* r1b   = (const float*)d_in[8];
    const float* r2w   = (const float*)d_in[9];
    const float* r2root= (const float*)d_in[10];
    const float* r2b   = (const float*)d_in[11];
    const float* fc2W  = (const float*)d_in[12];
    const float* fc2b  = (const float*)d_in[13];
    const float* outW  = (const float*)d_in[14];
    const float* outb  = (const float*)d_in[15];
    float* out = (float*)d_out;

    // workspace layout (floats)
    float* ws = (float*)d_ws;
    size_t off = 0;
    float* h1   = ws + off; off += (size_t)NODES * FIN;   // fc1 output [N,768]
    float* bufa = ws + off; off += (size_t)NODES * FHID;  // layer1 output [N,256]
    float* hr0  = ws + off; off += (size_t)NODES * FHID;  // relation-0 transform
    float* hr1  = ws + off; off += (size_t)NODES * FHID;  // relation-1 transform
    float* cnt  = ws + off; off += (size_t)2 * NODES;     // per-relation inverse counts
    float* bufb = h1;                                     // reuse (h1 dead after layer1)
    float* h3   = h1 + (size_t)NODES * FHID;              // reuse (fc2 output)

    // edge-type counts (identical for both RGCN layers) -> inverse means
    fill_zero_kernel<<<(2 * NODES + 255) / 256, 256, 0, stream>>>(cnt, 2 * NODES);
    count_edges_kernel<<<(EDGES + 255) / 256, 256, 0, stream>>>(edst, etype, cnt, EDGES, NODES);
    invert_cnt_kernel<<<(2 * NODES + 255) / 256, 256, 0, stream>>>(cnt, 2 * NODES);

    auto gemm = [&](const float* A, const float* W, const float* b, float* Cm,
                    int M, int K, int N, int leaky) {
        int waves  = (M / 16) * (N / 64);
        int blocks = (waves + 7) / 8;   // 8 waves / 256-thread block
        wmma_gemm_f32<<<blocks, 256, 0, stream>>>(A, W, b, Cm, M, K, N, leaky);
    };

    const int scatter_blocks = (int)(((long)EDGES * 32 + 255) / 256);

    // h1 = leaky_relu(x @ fc1_W + fc1_b)
    gemm(x, fc1W, fc1b, h1, NODES, FIN, FIN, 1);

    // RGCN layer 1
    gemm(h1, r1root, r1b, bufa, NODES, FIN, FHID, 0);
    gemm(h1, r1w,                       nullptr, hr0, NODES, FIN, FHID, 0);
    gemm(h1, r1w + (size_t)FIN * FHID,  nullptr, hr1, NODES, FIN, FHID, 0);
    scatter_mean_kernel<<<scatter_blocks, 256, 0, stream>>>(hr0, hr1, esrc, edst, etype,
                                                            cnt, bufa, EDGES, NODES);

    // RGCN layer 2 (input bufa, output bufb which reuses h1's storage)
    gemm(bufa, r2root, r2b, bufb, NODES, FHID, FHID, 0);
    gemm(bufa, r2w,                        nullptr, hr0, NODES, FHID, FHID, 0);
    gemm(bufa, r2w + (size_t)FHID * FHID,  nullptr, hr1, NODES, FHID, FHID, 0);
    scatter_mean_kernel<<<scatter_blocks, 256, 0, stream>>>(hr0, hr1, esrc, edst, etype,
                                                            cnt, bufb, EDGES, NODES);

    // h3 = leaky_relu(bufb @ fc2_W + fc2_b)
    gemm(bufb, fc2W, fc2b, h3, NODES, FHID, FHID, 1);

    // softmax(h3 @ out_W + out_b)
    final_softmax_kernel<<<(NODES * 32 + 255) / 256, 256, 0, stream>>>(h3, outW, outb, out, NODES);
}